// KarateGraph4GCN_68599217652370
// MI455X (gfx1250) — compile-verified
//
#include <hip/hip_runtime.h>
#include <hip/hip_bf16.h>

typedef float v2f __attribute__((ext_vector_type(2)));
typedef float v8f __attribute__((ext_vector_type(8)));

#define GCN_N 10000
#define GCN_E 160000
#define KC 64   // K-chunk staged in LDS (16 rows x 64 cols = 4 KB per buffer)

// ---------------- degree / normalization ----------------

__global__ void deg_init(float* deg, int n) {
    int i = blockIdx.x * blockDim.x + threadIdx.x;
    if (i < n) deg[i] = 1.0f;  // self-loop contributes 1 to every node's degree
}

__global__ void deg_scatter(const int* __restrict__ dst, float* deg, int e) {
    int i = blockIdx.x * blockDim.x + threadIdx.x;
    if (i < e)
        (void)__hip_atomic_fetch_add(&deg[dst[i]], 1.0f,
                                     __ATOMIC_RELAXED, __HIP_MEMORY_SCOPE_AGENT);
}

__global__ void to_rsqrt(float* deg, int n) {
    int i = blockIdx.x * blockDim.x + threadIdx.x;
    if (i < n) deg[i] = rsqrtf(deg[i]);  // deg >= 1 always (self-loops)
}

// ---------------- fp32 WMMA GEMM: H = X @ W ----------------
// Block = 256 threads (8 waves), all waves share the 16-row A tile (same tm).
// A chunks are double-buffered through LDS via global_load_async_to_lds_b128
// (ASYNCcnt): chunk i+1's async copy overlaps WMMA work on chunk i.
// Each wave register-blocks a 16x64 strip (4 n-tiles, 4 v8f accumulators).
// FULL=true: every wave owns exactly 4 tiles (C % 512 == 0) -> no guards at all.
// FULL=false: tile count forced wave-uniform via readfirstlane -> scalar
// branches only, EXEC stays all-1s for every WMMA (ISA 7.12 requirement).
// B addressing uses 32-bit element offsets (W <= 2 MB) so loads are
// sgpr-base + 32-bit voffset with immediate t*16 displacements.
// A-frag (16x4 fp32, ISA 7.12.2): lane L holds row M = L&15, K = k0+2*(L>>4)+{0,1}.
// B-frag (4x16): K = k0 + 2*(L>>4) + j, N-col = L&15 (C/D-style striping).
// C/D: VGPR r -> row 16*tm + r + 8*(L>>4).
template <bool FULL>
__global__ void gemm_wmma_f32(const float* __restrict__ X, const float* __restrict__ W,
                              float* __restrict__ H, int M, int K, int NC) {
    __shared__ float atile[2][16 * KC];

    const int lane = threadIdx.x & 31;
    const int wave = threadIdx.x >> 5;
    const int tm = blockIdx.x;
    const int hi = lane >> 4;    // 0 or 1
    const int lo = lane & 15;

    const int tn0 = (blockIdx.y * 8 + wave) * 4;        // first n-tile of this wave
    int ntu = 4;
    if (!FULL) {
        const int ntv = (NC / 16) - tn0;                // tiles owned (may be <=0)
        ntu = __builtin_amdgcn_readfirstlane(ntv > 4 ? 4 : ntv);  // scalar
    }

    // async staging: thread t copies 16B = row (t>>4), cols 4*(t&15) .. +3
    const int srow = threadIdx.x >> 4;
    const int scol = (threadIdx.x & 15) * 4;
    const unsigned lds_st[2] = {
        (unsigned)(uintptr_t)(&atile[0][srow * KC + scol]),  // low 32 bits = LDS offset
        (unsigned)(uintptr_t)(&atile[1][srow * KC + scol])
    };
    unsigned long long gsrc =
        (unsigned long long)(uintptr_t)(X + (size_t)(tm * 16 + srow) * K + scol);

    v8f acc[4] = {};

    // 32-bit walking element offsets into W for rows k and k+1 (k = 2*hi at start).
    unsigned boff0 = (unsigned)(2 * hi) * (unsigned)NC + (unsigned)(tn0 * 16 + lo);
    unsigned boff1 = boff0 + (unsigned)NC;

    // prologue: kick off chunk 0
    asm volatile("global_load_async_to_lds_b128 %0, %1, off"
                 :: "v"(lds_st[0]), "v"(gsrc) : "memory");
    gsrc += (unsigned long long)KC * sizeof(float);

    int par = 0;
    for (int k0 = 0; k0 < K; k0 += KC) {
        asm volatile("s_wait_asynccnt 0x0" ::: "memory");  // chunk k0 landed (this wave)
        __syncthreads();                                   // ... and from all waves

        if (k0 + KC < K) {  // overlap: start next chunk into the other buffer
            asm volatile("global_load_async_to_lds_b128 %0, %1, off"
                         :: "v"(lds_st[par ^ 1]), "v"(gsrc) : "memory");
            gsrc += (unsigned long long)KC * sizeof(float);
        }

        if (FULL || ntu > 0) {
            const float2* ap =
                reinterpret_cast<const float2*>(&atile[par][lo * KC + 2 * hi]);
            #pragma unroll 4
            for (int kk = 0; kk < KC; kk += 4) {
                const float2 av = ap[kk >> 1];
                v2f a; a.x = av.x; a.y = av.y;
                #pragma unroll
                for (int t = 0; t < 4; ++t) {
                    if (FULL || t < ntu) {  // scalar guard (folds away when FULL)
                        v2f b; b.x = W[boff0 + t * 16]; b.y = W[boff1 + t * 16];
                        acc[t] = __builtin_amdgcn_wmma_f32_16x16x4_f32(
                            /*neg_a=*/false, a, /*neg_b=*/false, b,
                            /*c_mod=*/(short)0, acc[t],
                            /*reuse_a=*/false, /*reuse_b=*/false);
                    }
                }
                boff0 += 4u * (unsigned)NC;   // k advances by 4
                boff1 += 4u * (unsigned)NC;
            }
        }
        par ^= 1;
    }

    #pragma unroll
    for (int t = 0; t < 4; ++t) {
        if (FULL || t < ntu) {
            const int n = (tn0 + t) * 16 + lo;
            float* orow = H + (size_t)(tm * 16 + 8 * hi) * NC + n;
            #pragma unroll
            for (int r = 0; r < 8; ++r) orow[(size_t)r * NC] = acc[t][r];
        }
    }
}

// ---------------- aggregation ----------------

// self-loop: out[i] = h[i] * dinv[i]^2  (also serves as the zero-free init)
__global__ void selfloop_init(const float* __restrict__ h, const float* __restrict__ dinv,
                              float* __restrict__ out, long long total, int C) {
    long long idx = (long long)blockIdx.x * blockDim.x + threadIdx.x;
    if (idx >= total) return;
    int i = (int)(idx / C);
    float d = dinv[i];
    out[idx] = h[idx] * d * d;
}

// one wave per edge: out[dst] += h[src] * dinv[src]*dinv[dst]
__global__ void edge_scatter(const int* __restrict__ src, const int* __restrict__ dst,
                             const float* __restrict__ h, const float* __restrict__ dinv,
                             float* __restrict__ out, int e, int C) {
    int w = blockIdx.x * (blockDim.x >> 5) + (threadIdx.x >> 5);
    if (w >= e) return;
    int lane = threadIdx.x & 31;
    int s = src[w], d = dst[w];
    float nrm = dinv[s] * dinv[d];
    const float* hs = h + (size_t)s * C;
    float* od = out + (size_t)d * C;
    for (int c = lane; c < C; c += 32)
        (void)__hip_atomic_fetch_add(&od[c], hs[c] * nrm,
                                     __ATOMIC_RELAXED, __HIP_MEMORY_SCOPE_AGENT);
}

__global__ void bias_act(float* __restrict__ a, const float* __restrict__ bias,
                         long long total, int C, int do_relu) {
    long long idx = (long long)blockIdx.x * blockDim.x + threadIdx.x;
    if (idx >= total) return;
    float v = a[idx] + bias[(int)(idx % C)];
    a[idx] = do_relu ? fmaxf(v, 0.0f) : v;
}

// ---------------- log_softmax over 32 columns (one wave32 per row) ----------------

__global__ void log_softmax32(const float* __restrict__ in, float* __restrict__ out, int n) {
    int lane = threadIdx.x & 31;
    int row = blockIdx.x * (blockDim.x >> 5) + (threadIdx.x >> 5);
    if (row >= n) return;
    float v = in[(size_t)row * 32 + lane];
    float m = v;
    #pragma unroll
    for (int o = 16; o > 0; o >>= 1) m = fmaxf(m, __shfl_xor(m, o, 32));
    float s = expf(v - m);
    #pragma unroll
    for (int o = 16; o > 0; o >>= 1) s += __shfl_xor(s, o, 32);
    out[(size_t)row * 32 + lane] = v - m - logf(s);
}

// ---------------- host launch ----------------

extern "C" void kernel_launch(void* const* d_in, const int* in_sizes, int n_in,
                              void* d_out, int out_size, void* d_ws, size_t ws_size,
                              hipStream_t stream) {
    const float* x  = (const float*)d_in[0];
    const int*   ei = (const int*)d_in[1];     // [2, E] flattened: row0=src, row1=dst
    const float* W1 = (const float*)d_in[2]; const float* b1 = (const float*)d_in[3];
    const float* W2 = (const float*)d_in[4]; const float* b2 = (const float*)d_in[5];
    const float* W3 = (const float*)d_in[6]; const float* b3 = (const float*)d_in[7];
    const float* W4 = (const float*)d_in[8]; const float* b4 = (const float*)d_in[9];

    const int N = GCN_N, E = GCN_E;
    const int* src = ei;
    const int* dst = ei + E;

    float* ws   = (float*)d_ws;
    float* dinv = ws;                                  // N floats
    float* bufA = ws + 10240;                          // N*1024 floats (GEMM out)
    float* bufB = bufA + (size_t)N * 1024;             // N*1024 floats (aggregated act)

    // normalization: deg (with self-loop) -> deg^{-1/2}
    deg_init   <<<(N + 255) / 256, 256, 0, stream>>>(dinv, N);
    deg_scatter<<<(E + 255) / 256, 256, 0, stream>>>(dst, dinv, E);
    to_rsqrt   <<<(N + 255) / 256, 256, 0, stream>>>(dinv, N);

    auto run_layer = [&](const float* act, const float* Wm, const float* bias,
                         int K, int C, bool relu) {
        // each block covers 16 rows x 512 cols (8 waves x 4 n-tiles)
        dim3 gg(N / 16, (C + 511) / 512);
        if (C % 512 == 0)
            gemm_wmma_f32<true><<<gg, 256, 0, stream>>>(act, Wm, bufA, N, K, C);
        else
            gemm_wmma_f32<false><<<gg, 256, 0, stream>>>(act, Wm, bufA, N, K, C);
        long long total = (long long)N * C;
        int eb = (int)((total + 255) / 256);
        selfloop_init<<<eb, 256, 0, stream>>>(bufA, dinv, bufB, total, C);
        edge_scatter <<<(E + 7) / 8, 256, 0, stream>>>(src, dst, bufA, dinv, bufB, E, C);
        bias_act     <<<eb, 256, 0, stream>>>(bufB, bias, total, C, relu ? 1 : 0);
    };

    run_layer(x,    W1, b1,  128,  512, true);
    run_layer(bufB, W2, b2,  512, 1024, true);
    run_layer(bufB, W3, b3, 1024,  512, true);
    run_layer(bufB, W4, b4,  512,   32, false);

    log_softmax32<<<(N + 7) / 8, 256, 0, stream>>>(bufB, (float*)d_out, N);
}